// Gcn_24395414242126
// MI455X (gfx1250) — compile-verified
//
#include <hip/hip_runtime.h>

#define D 128
#define APAD 132   // LDS row pitch in floats: 132*4B = 528B -> bank stride 4, conflict-free b64 reads

typedef float v2f __attribute__((ext_vector_type(2)));
typedef float v8f __attribute__((ext_vector_type(8)));

// LDS byte-offset of a generic pointer into shared memory (for async-to-LDS VDST operand)
__device__ __forceinline__ uint32_t lds_off(const void* p) {
    return (uint32_t)(uintptr_t)(__attribute__((address_space(3))) const char*)p;
}
// CDNA5 async global->LDS copy, 16B per lane, tracked by ASYNCcnt
__device__ __forceinline__ void async_g2l_b128(uint32_t lds_byte, const void* gaddr) {
    asm volatile("global_load_async_to_lds_b128 %0, %1, off" :: "v"(lds_byte), "v"(gaddr) : "memory");
}
__device__ __forceinline__ void wait_async0() {
    asm volatile("s_wait_asynccnt 0x0" ::: "memory");
}

// ---------------- utility zero kernels ----------------
__global__ void k_zero_f(float* p, int n) {
    int i = blockIdx.x * blockDim.x + threadIdx.x;
    if (i < n) p[i] = 0.0f;
}
__global__ void k_zero_i(int* p, int n) {
    int i = blockIdx.x * blockDim.x + threadIdx.x;
    if (i < n) p[i] = 0;
}

// ---------------- degree histogram over destinations ----------------
__global__ void k_count(const int* __restrict__ dst, int* deg, int E) {
    int e = blockIdx.x * blockDim.x + threadIdx.x;
    if (e < E) atomicAdd(&deg[dst[e]], 1);
}

// dinv[n] = rsqrt(deg_in[n] + 1 self-loop); deg >= 1 always
__global__ void k_dinv(const int* __restrict__ deg, float* dinv, int N) {
    int i = blockIdx.x * blockDim.x + threadIdx.x;
    if (i < N) dinv[i] = rsqrtf((float)(deg[i] + 1));
}

// ---------------- single-block exclusive scan (N = 100k) ----------------
__global__ void k_scan(const int* __restrict__ deg, int* offs, int n) {
    __shared__ int tmp[1024];
    __shared__ int carry;
    if (threadIdx.x == 0) carry = 0;
    __syncthreads();
    for (int base = 0; base < n; base += 1024) {
        int i = base + threadIdx.x;
        int v = (i < n) ? deg[i] : 0;
        tmp[threadIdx.x] = v;
        __syncthreads();
        for (int off = 1; off < 1024; off <<= 1) {
            int t = (threadIdx.x >= off) ? tmp[threadIdx.x - off] : 0;
            __syncthreads();
            tmp[threadIdx.x] += t;
            __syncthreads();
        }
        if (i < n) offs[i] = carry + tmp[threadIdx.x] - v;
        int total = tmp[1023];
        __syncthreads();
        if (threadIdx.x == 0) carry += total;
        __syncthreads();
    }
    if (threadIdx.x == 0) offs[n] = carry;
}

// ---------------- CSR bucket fill ----------------
__global__ void k_fill(const int* __restrict__ src, const int* __restrict__ dst,
                       const int* __restrict__ offs, int* cursor, int* srclist, int E) {
    int e = blockIdx.x * blockDim.x + threadIdx.x;
    if (e < E) {
        int d = dst[e];
        int p = offs[d] + atomicAdd(&cursor[d], 1);
        srclist[p] = src[e];
    }
}

// ---------------- FP32 WMMA GEMM: out[M x 128] = A[M x 128] @ W[128 x 128] (+bias) ----
// Block = 256 threads = 8 waves; each wave owns a 16-row M strip, all 8 N-tiles.
// A tile staged in LDS via global_load_async_to_lds_b128 (coalesced, ASYNCcnt-tracked).
// W staged transposed in LDS so every B fragment is one contiguous ds_load_b64.
__global__ __launch_bounds__(256) void gemm_wmma_f32(
        const float* __restrict__ A, const float* __restrict__ W,
        const float* __restrict__ bias, float* __restrict__ out, int M) {
    extern __shared__ float smem[];
    float* wt = smem;               // transposed W: wt[c*APAD + k] = W[k][c]
    float* at = smem + D * APAD;    // A tile:      at[r*APAD + c]
    int tid = threadIdx.x;
    int rowbase0 = blockIdx.x * 128;

    // ---- stage A tile (async, 16B/lane, rows clamped instead of predicated) ----
    for (int idx = tid * 4; idx < D * D; idx += 256 * 4) {
        int r = idx >> 7;
        int c = idx & (D - 1);
        int grow = min(rowbase0 + r, M - 1);   // clamp: dup rows feed only unstored outputs
        async_g2l_b128(lds_off(at + r * APAD + c), A + (size_t)grow * D + c);
    }
    // ---- stage W transposed (coalesced global float4 read, scalar LDS stores) ----
    for (int idx = tid * 4; idx < D * D; idx += 256 * 4) {
        int r = idx >> 7;
        int c = idx & (D - 1);
        float4 v = *(const float4*)(W + idx);
        wt[(c + 0) * APAD + r] = v.x;
        wt[(c + 1) * APAD + r] = v.y;
        wt[(c + 2) * APAD + r] = v.z;
        wt[(c + 3) * APAD + r] = v.w;
    }
    wait_async0();
    __syncthreads();

    int wave = tid >> 5;
    int lane = tid & 31;
    int m  = lane & 15;      // A row / B column within tile
    int kh = lane >> 4;      // half-wave K-pair select

    const float* arow = at + (wave * 16 + m) * APAD;
    v8f c8[8] = {};
    for (int k0 = 0; k0 < D; k0 += 4) {
        v2f a = *(const v2f*)(arow + k0 + 2 * kh);            // ds_load_b64, conflict-free
#pragma unroll
        for (int nt = 0; nt < 8; ++nt) {
            int col = nt * 16 + m;
            v2f b = *(const v2f*)(wt + col * APAD + k0 + 2 * kh); // ds_load_b64, conflict-free
            c8[nt] = __builtin_amdgcn_wmma_f32_16x16x4_f32(
                false, a, false, b, (short)0, c8[nt], false, false);
        }
    }

    // C/D layout: lane holds column n = lane&15; VGPR r -> row r + 8*(lane>>4)
    int n    = lane & 15;
    int rsel = lane >> 4;
#pragma unroll
    for (int r = 0; r < 8; ++r) {
        int row = rowbase0 + wave * 16 + r + 8 * rsel;
        if (row < M) {
#pragma unroll
            for (int nt = 0; nt < 8; ++nt) {
                int col = nt * 16 + n;
                float v = c8[nt][r];
                if (bias) v += bias[col];
                out[(size_t)row * D + col] = v;
            }
        }
    }
}

// ---------------- CSR gather + bias + ReLU + LDS-staged graph pooling ----------------
// 256 threads = 8 waves; block owns 64 consecutive nodes (8 per wave).
// batch ids are sorted, so a block spans 1-2 graphs -> flush only those rows.
#define NODES_PER_BLOCK 64
__global__ void k_gather_pool(const float* __restrict__ xw, const float* __restrict__ dinv,
                              const int* __restrict__ offs, const int* __restrict__ srclist,
                              const float* __restrict__ b1, const int* __restrict__ batch,
                              float* gsum, float* gcnt, int N) {
    __shared__ float lgs[64 * D];   // 32 KB local pool table (G <= 64)
    for (int i = threadIdx.x; i < 64 * D; i += blockDim.x) lgs[i] = 0.0f;
    __syncthreads();

    int wave = threadIdx.x >> 5;
    int lane = threadIdx.x & 31;
    int blockStart = blockIdx.x * NODES_PER_BLOCK;
    float4 bvec = *(const float4*)(b1 + lane * 4);

    for (int i = 0; i < 8; ++i) {
        int node = blockStart + wave * 8 + i;
        if (node >= N) break;                      // uniform per wave
        float di = dinv[node];
        float4 acc = make_float4(0.f, 0.f, 0.f, 0.f);
        int beg = offs[node], end = offs[node + 1];
        for (int e = beg; e < end; ++e) {
            int r = srclist[e];
            float nm = di * dinv[r];
            float4 v = *(const float4*)(xw + (size_t)r * D + lane * 4);
            acc.x += v.x * nm; acc.y += v.y * nm;
            acc.z += v.z * nm; acc.w += v.w * nm;
        }
        // self-loop term: norm = dinv[n]^2
        {
            float nm = di * di;
            float4 v = *(const float4*)(xw + (size_t)node * D + lane * 4);
            acc.x += v.x * nm; acc.y += v.y * nm;
            acc.z += v.z * nm; acc.w += v.w * nm;
        }
        // bias + ReLU
        acc.x = fmaxf(acc.x + bvec.x, 0.f);
        acc.y = fmaxf(acc.y + bvec.y, 0.f);
        acc.z = fmaxf(acc.z + bvec.z, 0.f);
        acc.w = fmaxf(acc.w + bvec.w, 0.f);

        int g = batch[node];
        float* p = lgs + g * D + lane * 4;
        atomicAdd(p + 0, acc.x);   // ds_add_f32
        atomicAdd(p + 1, acc.y);
        atomicAdd(p + 2, acc.z);
        atomicAdd(p + 3, acc.w);
        if (lane == 0) unsafeAtomicAdd(&gcnt[g], 1.0f);
    }
    __syncthreads();

    int nLast = min(blockStart + NODES_PER_BLOCK, N) - 1;
    int gmin = batch[blockStart];
    int gmax = batch[nLast];
    int cnt = (gmax - gmin + 1) * D;
    for (int i = threadIdx.x; i < cnt; i += blockDim.x) {
        float v = lgs[gmin * D + i];
        if (v != 0.0f) unsafeAtomicAdd(&gsum[gmin * D + i], v);
    }
}

// ---------------- mean pool finalize ----------------
__global__ void k_pool(const float* __restrict__ gsum, const float* __restrict__ gcnt,
                       float* pooled, int GD) {
    int i = blockIdx.x * blockDim.x + threadIdx.x;
    if (i < GD) {
        int g = i / D;
        pooled[i] = gsum[i] / fmaxf(gcnt[g], 1.0f);
    }
}

extern "C" void kernel_launch(void* const* d_in, const int* in_sizes, int n_in,
                              void* d_out, int out_size, void* d_ws, size_t ws_size,
                              hipStream_t stream) {
    const float* x     = (const float*)d_in[0];
    const int*   eidx  = (const int*)d_in[1];   // [2, E] row-major
    const int*   batch = (const int*)d_in[2];
    const float* W1 = (const float*)d_in[3];
    const float* b1 = (const float*)d_in[4];
    const float* Wl = (const float*)d_in[5];
    const float* bl = (const float*)d_in[6];

    int N = in_sizes[0] / D;
    int E = in_sizes[1] / 2;
    int G = out_size / D;
    const int* src = eidx;       // edge_index[0]
    const int* dst = eidx + E;   // edge_index[1]

    // carve workspace (256B-aligned regions)
    char* ws = (char*)d_ws;
    auto alloc = [&](size_t bytes) {
        char* p = ws;
        ws += (bytes + 255) & ~(size_t)255;
        return p;
    };
    float* xw      = (float*)alloc((size_t)N * D * sizeof(float));
    float* dinv    = (float*)alloc((size_t)N * sizeof(float));
    int*   degc    = (int*)  alloc((size_t)N * sizeof(int));
    int*   offs    = (int*)  alloc((size_t)(N + 1) * sizeof(int));
    int*   cursor  = (int*)  alloc((size_t)N * sizeof(int));
    int*   srclist = (int*)  alloc((size_t)E * sizeof(int));
    float* gsum    = (float*)alloc((size_t)G * D * sizeof(float));
    float* gcnt    = (float*)alloc((size_t)G * sizeof(float));
    float* pooled  = (float*)alloc((size_t)G * D * sizeof(float));
    (void)ws_size; (void)n_in;

    const int GEMM_SMEM = 2 * D * APAD * (int)sizeof(float);   // 132 KB of 320 KB WGP LDS
    static bool attr_done = false;
    (void)hipFuncSetAttribute((const void*)gemm_wmma_f32,
                              hipFuncAttributeMaxDynamicSharedMemorySize, GEMM_SMEM);
    (void)attr_done;

    int nbN = (N + 255) / 256;
    int nbE = (E + 255) / 256;

    k_zero_i<<<nbN, 256, 0, stream>>>(degc, N);
    k_zero_i<<<nbN, 256, 0, stream>>>(cursor, N);
    k_zero_f<<<(G * D + 255) / 256, 256, 0, stream>>>(gsum, G * D);
    k_zero_f<<<1, 256, 0, stream>>>(gcnt, G);

    k_count<<<nbE, 256, 0, stream>>>(dst, degc, E);
    k_dinv<<<nbN, 256, 0, stream>>>(degc, dinv, N);
    k_scan<<<1, 1024, 0, stream>>>(degc, offs, N);
    k_fill<<<nbE, 256, 0, stream>>>(src, dst, offs, cursor, srclist, E);

    // xw = x @ W1 (FP32 WMMA, LDS-staged via async copy)
    gemm_wmma_f32<<<(N + 127) / 128, 256, GEMM_SMEM, stream>>>(x, W1, nullptr, xw, N);

    // CSR gather + bias + relu + pooled sums
    k_gather_pool<<<(N + NODES_PER_BLOCK - 1) / NODES_PER_BLOCK, 256, 0, stream>>>(
        xw, dinv, offs, srclist, b1, batch, gsum, gcnt, N);

    k_pool<<<(G * D + 255) / 256, 256, 0, stream>>>(gsum, gcnt, pooled, G * D);

    // final: out = pooled @ Wl + bl (FP32 WMMA, single block)
    gemm_wmma_f32<<<1, 256, GEMM_SMEM, stream>>>(pooled, Wl, bl, (float*)d_out, G);
}